// GeoAttentionBias_50878182589111
// MI455X (gfx1250) — compile-verified
//
#include <hip/hip_runtime.h>

typedef __attribute__((ext_vector_type(2))) float v2f;
typedef __attribute__((ext_vector_type(8))) float v8f;

#define NB   32      // blades
#define SEQ  1024    // sequence length
#define NH   16      // heads
#define IB   128     // i-rows per workgroup
#define JBLK 128     // j-rows per staged block
#define NJB  2       // staged j-blocks per workgroup (grid.y = 4 -> covers 1024)
#define PAD  33      // LDS row pitch (floats) to avoid bank conflicts
#define KBUF (JBLK * PAD)

// ---------------------------------------------------------------------------
// Kernel 1: W[h][a][b] = lambda * reorder_sign(a,b) * metric(a|b) * psi[h][a|b]
// (zero when a&b != 0).  Lambda folded in so the main GEMM has no epilogue.
// ---------------------------------------------------------------------------
__global__ void build_w_kernel(const float* __restrict__ psi,
                               const float* __restrict__ lam,
                               float* __restrict__ W) {
  const int h = blockIdx.x;          // 16 heads
  const int t = threadIdx.x;         // 1024 = 32*32
  const int a = t >> 5;
  const int b = t & 31;
  float val = 0.0f;
  if ((a & b) == 0) {
    const int k = a | b;
    int s = 0;
    int aa = a >> 1;
    while (aa) { s += __popc(aa & b); aa >>= 1; }
    float sign = (s & 1) ? -1.0f : 1.0f;
    if (k & 16) sign = -sign;        // metric: e4 squares to -1
    val = sign * psi[h * NB + k] * lam[0];
  }
  W[h * NB * NB + t] = val;
}

// ---------------------------------------------------------------------------
// Async staging: one 128x32 f32 block, global -> LDS, via
// GLOBAL_LOAD_ASYNC_TO_LDS_B128 (ASYNCcnt).  256 threads x 4 b128 each.
// gbase is workgroup-uniform (lands in an SGPR pair -> GVS addressing).
// ---------------------------------------------------------------------------
__device__ __forceinline__ void stage_async_block(const float* gbase,
                                                  float* lbuf, int tid) {
  const unsigned long long sbase = (unsigned long long)gbase;
#pragma unroll
  for (int it = 0; it < 4; ++it) {
    const int idx = tid + it * 256;        // float4 index, 1024 total
    const int row = idx >> 3;              // 8 float4 per 32-f32 row
    const int q4  = (idx & 7) * 4;
    const unsigned voff  = (unsigned)(idx * 16);   // contiguous global bytes
    const unsigned laddr =
        (unsigned)(unsigned long long)(lbuf + row * PAD + q4);
    asm volatile("global_load_async_to_lds_b128 %0, %1, %2"
                 :
                 : "v"(laddr), "v"(voff), "s"(sbase)
                 : "memory");
  }
}

// ---------------------------------------------------------------------------
// Kernel 2: bias[z,h,i,j] = sum_b (Q W)[i,b] * K[j,b]  via f32 WMMA.
//
// WMMA f32 16x16x4 operand layouts (wave32):
//   A (16x4): lane%16 = row M, VGPR v + 2*(lane/16) = K
//   B (4x16): lane%16 = col N, VGPR v + 2*(lane/16) = K
//   D (16x16): lane%16 = col N, VGPR v + 8*(lane/16) = row M
// ---------------------------------------------------------------------------
__launch_bounds__(256)
__global__ void geo_bias_kernel(const float* __restrict__ Q,
                                const float* __restrict__ Kl,
                                const float* __restrict__ Wg,
                                float* __restrict__ out) {
  __shared__ float w_s [NB  * PAD];
  __shared__ float q_s [IB  * PAD];
  __shared__ float qw_s[IB  * PAD];
  __shared__ float k_s [2 * KBUF];         // double-buffered K blocks

  const int tid   = threadIdx.x;
  const int wave  = tid >> 5;        // 0..7
  const int lane  = tid & 31;
  const int lhalf = lane >> 4;       // 0/1
  const int lmod  = lane & 15;

  const int bx   = blockIdx.x;       // (z*16+h)*8 + iblk  -> 256 blocks
  const int iblk = bx & 7;
  const int zh   = bx >> 3;          // 0..31
  const int h    = zh & 15;
  const int i0   = iblk * IB;

  const size_t qkbase = (size_t)zh * SEQ * NB;
  const size_t obase  = (size_t)zh * SEQ * SEQ;
  const int    jbase  = blockIdx.y * (NJB * JBLK);

  // ---- kick off async staging of K block 0 (hidden behind QW phase) -----
  stage_async_block(Kl + qkbase + (size_t)jbase * NB, k_s, tid);

  // ---- stage W (this head) into LDS --------------------------------------
  for (int idx = tid; idx < NB * NB; idx += 256)
    w_s[(idx >> 5) * PAD + (idx & 31)] = Wg[h * NB * NB + idx];

  // ---- stage Q block (128 rows x 32 f32) with b128 loads -----------------
  {
    const float4* qv = (const float4*)(Q + qkbase + (size_t)i0 * NB);
    for (int idx = tid; idx < IB * NB / 4; idx += 256) {
      const int row = idx >> 3;      // 8 float4 per row
      const int q4  = (idx & 7) * 4;
      const float4 v = qv[idx];
      q_s[row * PAD + q4 + 0] = v.x;
      q_s[row * PAD + q4 + 1] = v.y;
      q_s[row * PAD + q4 + 2] = v.z;
      q_s[row * PAD + q4 + 3] = v.w;
    }
  }
  __syncthreads();

  // ---- QW = Q * W (tiny: 128x32x32 FMAs, VALU) ---------------------------
  {
    const int i  = tid >> 1;          // 2 threads per row
    const int bs = (tid & 1) * 16;    // each covers 16 output blades
    float qr[NB];
#pragma unroll
    for (int a = 0; a < NB; ++a) qr[a] = q_s[i * PAD + a];
    for (int b = 0; b < 16; ++b) {
      float acc = 0.0f;
#pragma unroll
      for (int a = 0; a < NB; ++a)
        acc = fmaf(qr[a], w_s[a * PAD + bs + b], acc);
      qw_s[i * PAD + bs + b] = acc;
    }
  }
  __syncthreads();

  // ---- per-wave A fragments: fixed 16-row i-tile, loaded once ------------
  float afr[16];
  {
    const int arow = wave * 16 + lmod;
#pragma unroll
    for (int c = 0; c < 8; ++c) {
      afr[2 * c + 0] = qw_s[arow * PAD + 4 * c + 2 * lhalf + 0];
      afr[2 * c + 1] = qw_s[arow * PAD + 4 * c + 2 * lhalf + 1];
    }
  }

#pragma unroll
  for (int jb = 0; jb < NJB; ++jb) {
    const int j0 = jbase + jb * JBLK;
    // prefetch next K block into the other buffer, then drain only the
    // older 4 async loads (async loads complete in order: wait cnt<=4).
    if (jb + 1 < NJB) {
      stage_async_block(Kl + qkbase + (size_t)(j0 + JBLK) * NB,
                        k_s + ((jb + 1) & 1) * KBUF, tid);
      asm volatile("s_wait_asynccnt 0x4" ::: "memory");
    } else {
      asm volatile("s_wait_asynccnt 0x0" ::: "memory");
    }
    __syncthreads();                 // async LDS writes visible to all waves

    const float* kb = k_s + (jb & 1) * KBUF;

    // ---- 8 j-tiles of 16 columns; 8 WMMA steps each (K = 32 blades) ------
    for (int jt = 0; jt < 8; ++jt) {
      const int jj = jt * 16;
      v8f acc = {0.f, 0.f, 0.f, 0.f, 0.f, 0.f, 0.f, 0.f};
#pragma unroll
      for (int c = 0; c < 8; ++c) {
        v2f a;
        a[0] = afr[2 * c + 0];
        a[1] = afr[2 * c + 1];
        const int krow = (jj + lmod) * PAD + 4 * c + 2 * lhalf;
        v2f b;
        b[0] = kb[krow + 0];
        b[1] = kb[krow + 1];
        acc = __builtin_amdgcn_wmma_f32_16x16x4_f32(
            /*neg_a=*/false, a, /*neg_b=*/false, b,
            /*c_mod=*/(short)0, acc, /*reuse_a=*/false, /*reuse_b=*/false);
      }
      // ---- store D tile: per VGPR, half-wave writes 64B contiguous -------
      const int col   = j0 + jj + lmod;
      const int rbase = i0 + wave * 16 + 8 * lhalf;
      float* op = out + obase + (size_t)rbase * SEQ + col;
#pragma unroll
      for (int v = 0; v < 8; ++v)
        op[(size_t)v * SEQ] = acc[v];
    }
  }
}

// ---------------------------------------------------------------------------
extern "C" void kernel_launch(void* const* d_in, const int* in_sizes, int n_in,
                              void* d_out, int out_size, void* d_ws, size_t ws_size,
                              hipStream_t stream) {
  const float* psi = (const float*)d_in[0];   // (16, 32)
  const float* Q   = (const float*)d_in[1];   // (2, 16, 1024, 32)
  const float* Kl  = (const float*)d_in[2];   // (2, 16, 1024, 32)
  const float* lam = (const float*)d_in[3];   // scalar
  float* out = (float*)d_out;                 // (2, 16, 1024, 1024)
  float* W   = (float*)d_ws;                  // 16*32*32 f32 = 64 KB scratch

  build_w_kernel<<<dim3(NH), dim3(NB * NB), 0, stream>>>(psi, lam, W);
  // grid.x: (z,h,i-block) = 2*16*8 = 256 ; grid.y: 4 j-ranges of 256 cols
  geo_bias_kernel<<<dim3(256, 4), dim3(256), 0, stream>>>(Q, Kl, W, out);
}